// ButterflyRotation_68994354643571
// MI455X (gfx1250) — compile-verified
//
#include <hip/hip_runtime.h>
#include <math.h>

#define DIM 1024
#define STAGES 10
#define NPAIR 512           // DIM/2 pairs per stage
#define RPB 8               // rows per block
#define TPB 256

// LDS word-index swizzle: all 2-way conflicts in our access patterns are pairs
// of words differing by exactly 64 (bit 6). XOR bits 3 and 5 (constant within
// each conflicting pattern, and they don't touch bits 0-1 so 16B blocks stay
// contiguous for b128 ops) when bit 6 is set -> every pattern conflict-free.
__device__ __forceinline__ int sw(int i) {
    return i ^ (((i >> 6) & 1) * 0x28);
}

__global__ void bf_build_tables(const float* __restrict__ angles,
                                float* __restrict__ ctab,
                                float* __restrict__ stab) {
    int tid = blockIdx.x * blockDim.x + threadIdx.x;
    if (tid < STAGES * NPAIR) {
        float s, c;
        sincosf(angles[tid], &s, &c);
        ctab[tid] = c;
        stab[tid] = s;
    }
}

__global__ __launch_bounds__(TPB) void bf_butterfly_kernel(
    const float* __restrict__ x,
    const float* __restrict__ ctab,
    const float* __restrict__ stab,
    float* __restrict__ out)
{
    __shared__ __align__(16) float smem[RPB][DIM];
    const int t = threadIdx.x;
    const long long row0 = (long long)blockIdx.x * RPB;

    // ---- CDNA5 async DMA: global -> LDS, 16B per lane per row ----
    const unsigned voff  = (unsigned)t * 16u;               // byte offset in row
    const unsigned ldst0 = (unsigned)(sw(4 * t)) * 4u;      // swizzled LDS byte addr
    #pragma unroll
    for (int r = 0; r < RPB; ++r) {
        const float* gbase = x + (row0 + r) * DIM;          // uniform -> SGPR pair
        unsigned ldst = ldst0 + (unsigned)(r * DIM) * 4u;
        asm volatile("global_load_async_to_lds_b128 %0, %1, %2"
                     :: "v"(ldst), "v"(voff), "s"(gbase) : "memory");
    }

    // ---- group 0: stages 0,1 in registers (thread owns elems 4t..4t+3) ----
    // Table loads issued before any async wait so they overlap the DMA.
    {
        const int ka = 2 * t, kb = 2 * t + 1;
        const float c0a = ctab[0 * NPAIR + ka], n0a = stab[0 * NPAIR + ka];
        const float c0b = ctab[0 * NPAIR + kb], n0b = stab[0 * NPAIR + kb];
        const float c1a = ctab[1 * NPAIR + ka], n1a = stab[1 * NPAIR + ka];
        const float c1b = ctab[1 * NPAIR + kb], n1b = stab[1 * NPAIR + kb];
        const int base = sw(4 * t);
        #pragma unroll
        for (int r = 0; r < RPB; ++r) {
            // async loads complete in order: row r ready once asynccnt <= RPB-1-r.
            // Group 0 reads only this thread's own 16B, so no barrier needed here.
            asm volatile("s_wait_asynccnt %0" :: "i"(RPB - 1 - r) : "memory");
            float4 v = *(const float4*)&smem[r][base];
            // stage 0: pairs (e0,e1),(e2,e3)
            float e0 = c0a * v.x - n0a * v.y;
            float e1 = n0a * v.x + c0a * v.y;
            float e2 = c0b * v.z - n0b * v.w;
            float e3 = n0b * v.z + c0b * v.w;
            // stage 1: pairs (e0,e2),(e1,e3)
            float u0 = c1a * e0 - n1a * e2;
            float u2 = n1a * e0 + c1a * e2;
            float u1 = c1b * e1 - n1b * e3;
            float u3 = n1b * e1 + c1b * e3;
            *(float4*)&smem[r][base] = make_float4(u0, u1, u2, u3);
        }
    }
    __syncthreads();

    // ---- groups 1..4: stages (2g, 2g+1); thread owns i = H<<(2g+2) | c<<2g | L ----
    #pragma unroll
    for (int g = 1; g <= 4; ++g) {
        const int s0 = 2 * g, s1 = 2 * g + 1;
        const int q  = 1 << (2 * g);            // 4^g = stride of stage s0
        const int L  = t & (q - 1);
        const int H  = t >> (2 * g);
        const int ka = (H << (2 * g + 1)) | L;  // pair idx for (e0,e1) and (e0,e2)
        const int kb = ka + q;                  // pair idx for (e2,e3) and (e1,e3)
        const float c0a = ctab[s0 * NPAIR + ka], n0a = stab[s0 * NPAIR + ka];
        const float c0b = ctab[s0 * NPAIR + kb], n0b = stab[s0 * NPAIR + kb];
        const float c1a = ctab[s1 * NPAIR + ka], n1a = stab[s1 * NPAIR + ka];
        const float c1b = ctab[s1 * NPAIR + kb], n1b = stab[s1 * NPAIR + kb];
        const int i0 = (H << (2 * g + 2)) | L;
        const int a0 = sw(i0), a1 = sw(i0 + q), a2 = sw(i0 + 2 * q), a3 = sw(i0 + 3 * q);
        #pragma unroll
        for (int r = 0; r < RPB; ++r) {
            float v0 = smem[r][a0], v1 = smem[r][a1];
            float v2 = smem[r][a2], v3 = smem[r][a3];
            // stage s0: (v0,v1) with ka, (v2,v3) with kb
            float e0 = c0a * v0 - n0a * v1;
            float e1 = n0a * v0 + c0a * v1;
            float e2 = c0b * v2 - n0b * v3;
            float e3 = n0b * v2 + c0b * v3;
            // stage s1: (e0,e2) with ka, (e1,e3) with kb
            float u0 = c1a * e0 - n1a * e2;
            float u2 = n1a * e0 + c1a * e2;
            float u1 = c1b * e1 - n1b * e3;
            float u3 = n1b * e1 + c1b * e3;
            if (g < 4) {
                smem[r][a0] = u0; smem[r][a1] = u1;
                smem[r][a2] = u2; smem[r][a3] = u3;
            } else {
                // g==4: i0 = t -> stores at t, t+256, t+512, t+768, coalesced.
                // Streamed output never re-read: non-temporal to spare L2.
                float* o = out + (row0 + r) * DIM;
                __builtin_nontemporal_store(u0, &o[i0]);
                __builtin_nontemporal_store(u1, &o[i0 + q]);
                __builtin_nontemporal_store(u2, &o[i0 + 2 * q]);
                __builtin_nontemporal_store(u3, &o[i0 + 3 * q]);
            }
        }
        if (g < 4) __syncthreads();
    }
}

extern "C" void kernel_launch(void* const* d_in, const int* in_sizes, int n_in,
                              void* d_out, int out_size, void* d_ws, size_t ws_size,
                              hipStream_t stream) {
    (void)in_sizes; (void)n_in; (void)ws_size;
    const float* x      = (const float*)d_in[0];
    const float* angles = (const float*)d_in[1];
    float* out  = (float*)d_out;
    float* ctab = (float*)d_ws;                 // 5120 floats
    float* stab = ctab + STAGES * NPAIR;        // 5120 floats (40KB total)

    bf_build_tables<<<(STAGES * NPAIR + 255) / 256, 256, 0, stream>>>(angles, ctab, stab);

    const int rows   = out_size / DIM;          // 65536
    const int blocks = rows / RPB;              // 8192
    bf_butterfly_kernel<<<blocks, TPB, 0, stream>>>(x, ctab, stab, out);
}